// Global_Confidence_Attention_40089224741203
// MI455X (gfx1250) — compile-verified
//
#include <hip/hip_runtime.h>

#define MODS  3
#define BATCH 8
#define SEQ   512
#define DIM   768
#define NORM_F 0.03608439182435161f              // 1/sqrt(768)
#define LOSS_SCALE (1.0f / 6291456.0f)           // 1/(M*B*S*S)

typedef float v2f __attribute__((ext_vector_type(2)));
typedef float v8f __attribute__((ext_vector_type(8)));

__device__ __forceinline__ v8f wmma_f32_16x16x4(v2f a, v2f b, v8f c) {
  // (neg_a, A, neg_b, B, c_mod, C, reuse_a, reuse_b)
  return __builtin_amdgcn_wmma_f32_16x16x4_f32(false, a, false, b, (short)0, c,
                                               false, false);
}

__device__ __forceinline__ void zero_acc(v8f acc[2][4]) {
#pragma unroll
  for (int fi = 0; fi < 2; ++fi)
#pragma unroll
    for (int fj = 0; fj < 4; ++fj)
#pragma unroll
      for (int r = 0; r < 8; ++r) acc[fi][fj][r] = 0.0f;
}

// One wave computes a 32x64 f32 tile (2x4 fragments of 16x16), K-looped in steps of 4.
// BT=1: B given as Bt[N,K] row-major (C = A @ B^T) -> contiguous b64 fragment loads.
// BT=0: B given as B[K,N] row-major (C = A @ B)    -> two b32 loads, ldb apart.
template <int BT>
__device__ __forceinline__ void gemm_wave(const float* __restrict__ A, int lda,
                                          const float* __restrict__ Bm, int ldb,
                                          int row0, int col0, int K,
                                          int l15, int half, v8f acc[2][4]) {
#pragma unroll 4
  for (int k0 = 0; k0 < K; k0 += 4) {
    v2f a[2], b[4];
#pragma unroll
    for (int fi = 0; fi < 2; ++fi) {
      const float* p = A + (size_t)(row0 + 16 * fi + l15) * lda + (k0 + 2 * half);
      a[fi].x = p[0];
      a[fi].y = p[1];
    }
#pragma unroll
    for (int fj = 0; fj < 4; ++fj) {
      if (BT) {
        const float* p = Bm + (size_t)(col0 + 16 * fj + l15) * ldb + (k0 + 2 * half);
        b[fj].x = p[0];
        b[fj].y = p[1];
      } else {
        const float* p = Bm + (size_t)(k0 + 2 * half) * ldb + (col0 + 16 * fj + l15);
        b[fj].x = p[0];
        b[fj].y = p[ldb];
      }
    }
#pragma unroll
    for (int fi = 0; fi < 2; ++fi)
#pragma unroll
      for (int fj = 0; fj < 4; ++fj)
        acc[fi][fj] = wmma_f32_16x16x4(a[fi], b[fj], acc[fi][fj]);
  }
}

// ---------------- Stage 1: proj = relu(X @ Wp^T + bp) ----------------
__global__ __launch_bounds__(256) void k_proj(const float* __restrict__ mod,
                                              const float* __restrict__ Wp,
                                              const float* __restrict__ bp,
                                              float* __restrict__ proj) {
  const int mb = blockIdx.x;  // m*BATCH + b
  const int m = mb / BATCH;
  const int wid = threadIdx.x >> 5, lane = threadIdx.x & 31;
  const int l15 = lane & 15, half = lane >> 4;
  const int row0 = blockIdx.y * 64 + (wid >> 2) * 32;
  const int col0 = blockIdx.z * 256 + (wid & 3) * 64;
  const float* A  = mod + (size_t)mb * SEQ * DIM;
  const float* Bt = Wp + (size_t)m * DIM * DIM;
  v8f acc[2][4];
  zero_acc(acc);
  gemm_wave<1>(A, DIM, Bt, DIM, row0, col0, DIM, l15, half, acc);
  float* C = proj + (size_t)mb * SEQ * DIM;
#pragma unroll
  for (int fj = 0; fj < 4; ++fj) {
    const int col = col0 + 16 * fj + l15;
    const float bias = bp[m * DIM + col];
#pragma unroll
    for (int fi = 0; fi < 2; ++fi) {
      const int rbase = row0 + 16 * fi + 8 * half;
#pragma unroll
      for (int r = 0; r < 8; ++r)
        C[(size_t)(rbase + r) * DIM + col] = fmaxf(acc[fi][fj][r] + bias, 0.0f);
    }
  }
}

// ---------------- Stage 2: scores_raw = relu((Pi @ Pj^T) * norm) ----------------
__global__ __launch_bounds__(256) void k_scores(const float* __restrict__ proj,
                                                float* __restrict__ scores) {
  const int p = blockIdx.x;  // (i*MODS + j)*BATCH + b
  const int b = p % BATCH;
  const int ij = p / BATCH;
  const int j = ij % MODS, i = ij / MODS;
  const int wid = threadIdx.x >> 5, lane = threadIdx.x & 31;
  const int l15 = lane & 15, half = lane >> 4;
  const int row0 = blockIdx.y * 64 + (wid >> 2) * 32;
  const int col0 = blockIdx.z * 256 + (wid & 3) * 64;
  const float* A  = proj + (size_t)(i * BATCH + b) * SEQ * DIM;
  const float* Bt = proj + (size_t)(j * BATCH + b) * SEQ * DIM;
  v8f acc[2][4];
  zero_acc(acc);
  gemm_wave<1>(A, DIM, Bt, DIM, row0, col0, DIM, l15, half, acc);
  float* C = scores + (size_t)p * SEQ * SEQ;
#pragma unroll
  for (int fj = 0; fj < 4; ++fj) {
    const int col = col0 + 16 * fj + l15;
#pragma unroll
    for (int fi = 0; fi < 2; ++fi) {
      const int rbase = row0 + 16 * fi + 8 * half;
#pragma unroll
      for (int r = 0; r < 8; ++r)
        C[(size_t)(rbase + r) * SEQ + col] = fmaxf(acc[fi][fj][r] * NORM_F, 0.0f);
    }
  }
}

// ---------------- Stage 3: in-place row softmax over 512 columns ----------------
__global__ __launch_bounds__(256) void k_softmax(float* __restrict__ scores) {
  __shared__ float red[256];
  float* row = scores + (size_t)blockIdx.x * SEQ;
  const int t = threadIdx.x;
  const float a = row[t];
  const float c = row[t + 256];
  red[t] = fmaxf(a, c);
  __syncthreads();
  for (int s = 128; s > 0; s >>= 1) {
    if (t < s) red[t] = fmaxf(red[t], red[t + s]);
    __syncthreads();
  }
  const float mx = red[0];
  __syncthreads();
  const float ea = __expf(a - mx), ec = __expf(c - mx);
  red[t] = ea + ec;
  __syncthreads();
  for (int s = 128; s > 0; s >>= 1) {
    if (t < s) red[t] += red[t + s];
    __syncthreads();
  }
  const float inv = 1.0f / red[0];
  row[t] = ea * inv;
  row[t + 256] = ec * inv;
}

// ---------------- Stage 4: values = scores @ proj_j ----------------
__global__ __launch_bounds__(256) void k_values(const float* __restrict__ scores,
                                                const float* __restrict__ proj,
                                                float* __restrict__ values) {
  const int p = blockIdx.x;
  const int b = p % BATCH;
  const int j = (p / BATCH) % MODS;
  const int wid = threadIdx.x >> 5, lane = threadIdx.x & 31;
  const int l15 = lane & 15, half = lane >> 4;
  const int row0 = blockIdx.y * 64 + (wid >> 2) * 32;
  const int col0 = blockIdx.z * 256 + (wid & 3) * 64;
  const float* A  = scores + (size_t)p * SEQ * SEQ;
  const float* Bm = proj + (size_t)(j * BATCH + b) * SEQ * DIM;
  v8f acc[2][4];
  zero_acc(acc);
  gemm_wave<0>(A, SEQ, Bm, DIM, row0, col0, SEQ, l15, half, acc);
  float* C = values + (size_t)p * SEQ * DIM;
#pragma unroll
  for (int fj = 0; fj < 4; ++fj) {
    const int col = col0 + 16 * fj + l15;
#pragma unroll
    for (int fi = 0; fi < 2; ++fi) {
      const int rbase = row0 + 16 * fi + 8 * half;
#pragma unroll
      for (int r = 0; r < 8; ++r)
        C[(size_t)(rbase + r) * DIM + col] = acc[fi][fj][r];
    }
  }
}

// ---------------- Stage 5: confidence + gca_loss ----------------
__global__ __launch_bounds__(256) void k_conf_loss(const float* __restrict__ proj,
                                                   const float* __restrict__ Wc,
                                                   const float* __restrict__ bc,
                                                   const float* __restrict__ scores,
                                                   float* __restrict__ loss) {
  const int mb = blockIdx.x;  // m*BATCH + b
  const int m = mb / BATCH, b = mb % BATCH;
  const int wid = threadIdx.x >> 5, lane = threadIdx.x & 31;
  const int l15 = lane & 15, half = lane >> 4;
  const int row0 = blockIdx.y * 64 + (wid >> 2) * 32;
  const int col0 = blockIdx.z * 256 + (wid & 3) * 64;
  const float* A = proj + (size_t)mb * SEQ * DIM;
  v8f acc[2][4];
  zero_acc(acc);
  gemm_wave<1>(A, DIM, Wc, DIM, row0, col0, DIM, l15, half, acc);
  // attn_scores_sum[m,b,s,t] = sum_i scores[i,m,b,s,t]
  const float* s0 = scores + (size_t)((0 * MODS + m) * BATCH + b) * SEQ * SEQ;
  const float* s1 = scores + (size_t)((1 * MODS + m) * BATCH + b) * SEQ * SEQ;
  const float* s2 = scores + (size_t)((2 * MODS + m) * BATCH + b) * SEQ * SEQ;
  float local = 0.0f;
#pragma unroll
  for (int fj = 0; fj < 4; ++fj) {
    const int col = col0 + 16 * fj + l15;
    const float bias = bc[col];
#pragma unroll
    for (int fi = 0; fi < 2; ++fi) {
      const int rbase = row0 + 16 * fi + 8 * half;
#pragma unroll
      for (int r = 0; r < 8; ++r) {
        const size_t off = (size_t)(rbase + r) * SEQ + col;
        const float conf = 1.0f / (1.0f + __expf(-(acc[fi][fj][r] + bias)));
        const float d = (s0[off] + s1[off] + s2[off]) - conf;
        local += d * d;
      }
    }
  }
  __shared__ float red[256];
  red[threadIdx.x] = local;
  __syncthreads();
  for (int s = 128; s > 0; s >>= 1) {
    if (threadIdx.x < s) red[threadIdx.x] += red[threadIdx.x + s];
    __syncthreads();
  }
  if (threadIdx.x == 0) atomicAdd(loss, red[0] * LOSS_SCALE);
}

// ---------------- Stage 6: out = scores @ values ----------------
__global__ __launch_bounds__(256) void k_out(const float* __restrict__ scores,
                                             const float* __restrict__ values,
                                             float* __restrict__ out) {
  const int p = blockIdx.x;
  const int wid = threadIdx.x >> 5, lane = threadIdx.x & 31;
  const int l15 = lane & 15, half = lane >> 4;
  const int row0 = blockIdx.y * 64 + (wid >> 2) * 32;
  const int col0 = blockIdx.z * 256 + (wid & 3) * 64;
  const float* A  = scores + (size_t)p * SEQ * SEQ;
  const float* Bm = values + (size_t)p * SEQ * DIM;
  v8f acc[2][4];
  zero_acc(acc);
  gemm_wave<0>(A, SEQ, Bm, DIM, row0, col0, SEQ, l15, half, acc);
  float* C = out + (size_t)p * SEQ * DIM;
#pragma unroll
  for (int fj = 0; fj < 4; ++fj) {
    const int col = col0 + 16 * fj + l15;
#pragma unroll
    for (int fi = 0; fi < 2; ++fi) {
      const int rbase = row0 + 16 * fi + 8 * half;
#pragma unroll
      for (int r = 0; r < 8; ++r)
        C[(size_t)(rbase + r) * DIM + col] = acc[fi][fj][r];
    }
  }
}

extern "C" void kernel_launch(void* const* d_in, const int* in_sizes, int n_in,
                              void* d_out, int out_size, void* d_ws, size_t ws_size,
                              hipStream_t stream) {
  (void)in_sizes; (void)n_in; (void)out_size; (void)ws_size;
  const float* mod = (const float*)d_in[0];  // [M,B,S,D]
  const float* Wp  = (const float*)d_in[1];  // [M,D,D]
  const float* bp  = (const float*)d_in[2];  // [M,D]
  const float* Wc  = (const float*)d_in[3];  // [S,D]
  const float* bc  = (const float*)d_in[4];  // [S]
  float* out = (float*)d_out;                // [0]=loss, [1:]=out flat [M,M,B,S,D]

  float* proj   = (float*)d_ws;                                      // M*B*S*D
  float* scores = proj + (size_t)MODS * BATCH * SEQ * DIM;           // M*M*B*S*S
  float* values = scores + (size_t)MODS * MODS * BATCH * SEQ * SEQ;  // M*M*B*S*D

  const dim3 blk(256);
  k_proj<<<dim3(MODS * BATCH, SEQ / 64, DIM / 256), blk, 0, stream>>>(mod, Wp, bp, proj);
  k_scores<<<dim3(MODS * MODS * BATCH, SEQ / 64, SEQ / 256), blk, 0, stream>>>(proj, scores);
  k_softmax<<<dim3(MODS * MODS * BATCH * SEQ), blk, 0, stream>>>(scores);
  k_values<<<dim3(MODS * MODS * BATCH, SEQ / 64, DIM / 256), blk, 0, stream>>>(scores, proj, values);
  hipMemsetAsync(d_out, 0, sizeof(float), stream);
  k_conf_loss<<<dim3(MODS * BATCH, SEQ / 64, SEQ / 256), blk, 0, stream>>>(proj, Wc, bc, scores, out);
  k_out<<<dim3(MODS * MODS * BATCH, SEQ / 64, DIM / 256), blk, 0, stream>>>(scores, values, out + 1);
}